// GCNConv_87806311399690
// MI455X (gfx1250) — compile-verified
//
#include <hip/hip_runtime.h>
#include <hip/hip_bf16.h>

typedef __attribute__((ext_vector_type(2))) float v2f;
typedef __attribute__((ext_vector_type(8))) float v8f;

#define DIM 512

// ---------------- degree / norm kernels ----------------

__global__ void deg_init_kernel(float* __restrict__ deg, int n) {
    int i = blockIdx.x * blockDim.x + threadIdx.x;
    if (i < n) deg[i] = 1.0f;   // self-loop weight
}

__global__ void deg_acc_kernel(const long long* __restrict__ ei,
                               const float* __restrict__ ea,
                               float* __restrict__ deg, int E) {
    int e = blockIdx.x * blockDim.x + threadIdx.x;
    if (e < E) {
        int col = (int)ei[(size_t)E + e];   // target node
        atomicAdd(&deg[col], ea[e]);
    }
}

__global__ void dinv_kernel(float* __restrict__ deg, int n) {
    int i = blockIdx.x * blockDim.x + threadIdx.x;
    if (i < n) {
        float d = deg[i];
        deg[i] = (d > 0.0f) ? rsqrtf(d) : 0.0f;
    }
}

// ---------------- WMMA fp32 GEMM: H = X @ W ----------------
// One wave32 computes a 16x64 strip of H (4 accumulators), K-loop in steps of
// 4 using V_WMMA_F32_16X16X4_F32. The A fragment is loaded once per K-step and
// reused across the 4 N-tiles. Layouts per CDNA5 ISA 7.12.2:
//   A 16x4: lanes 0-15 hold K=k0,k0+1 ; lanes 16-31 hold K=k0+2,k0+3 (row M=lane%16)
//   B 4x16: lanes 0-15 hold rows K=k0,k0+1 ; lanes 16-31 rows K=k0+2,k0+3 (col N=lane%16)
//   C/D: VGPR v, lane l -> (M = v + 8*(l/16), N = l%16)

#define WMMA_F32(A, B, C) \
    __builtin_amdgcn_wmma_f32_16x16x4_f32(false, (A), false, (B), (short)0, (C), false, false)

__global__ __launch_bounds__(256) void gemm_wmma_kernel(
        const float* __restrict__ X,   // [M, 512] row-major
        const float* __restrict__ Wm,  // [512, 512] row-major
        float* __restrict__ H,         // [M, 512]
        int M) {
    const int ngroups = DIM / 64;     // 8 groups of four 16-wide N tiles
    int wave = (blockIdx.x * blockDim.x + threadIdx.x) >> 5;
    int lane = threadIdx.x & 31;
    int mt = wave / ngroups;
    int ng = wave - mt * ngroups;
    if (mt * 16 >= M) return;                 // wave-uniform exit; EXEC stays all-1s

    int half = lane >> 4;                     // 0: K pair 0-1, 1: K pair 2-3
    int li   = lane & 15;

    int arow_idx = mt * 16 + li;
    if (arow_idx >= M) arow_idx = M - 1;      // clamp (safety; M=10000 is 16-aligned)
    const float* arow  = X  + (size_t)arow_idx * DIM;
    const float* bbase = Wm + ng * 64 + li;

    v8f c0 = {}, c1 = {}, c2 = {}, c3 = {};
    for (int k0 = 0; k0 < DIM; k0 += 4) {
        int ka = k0 + 2 * half;
        v2f a;
        a.x = arow[ka];
        a.y = arow[ka + 1];
        const float* brow0 = bbase + (size_t)ka * DIM;
        const float* brow1 = brow0 + DIM;
        v2f b0, b1, b2, b3;
        b0.x = brow0[0];  b0.y = brow1[0];
        b1.x = brow0[16]; b1.y = brow1[16];
        b2.x = brow0[32]; b2.y = brow1[32];
        b3.x = brow0[48]; b3.y = brow1[48];
        c0 = WMMA_F32(a, b0, c0);
        c1 = WMMA_F32(a, b1, c1);
        c2 = WMMA_F32(a, b2, c2);
        c3 = WMMA_F32(a, b3, c3);
    }

    float* hout = H + (size_t)(mt * 16) * DIM + ng * 64 + li;
    if (mt * 16 + 16 <= M) {                  // wave-uniform full-tile fast path
#pragma unroll
        for (int v = 0; v < 8; ++v) {
            size_t roff = (size_t)(v + 8 * half) * DIM;
            hout[roff +  0] = c0[v];
            hout[roff + 16] = c1[v];
            hout[roff + 32] = c2[v];
            hout[roff + 48] = c3[v];
        }
    } else {
#pragma unroll
        for (int v = 0; v < 8; ++v) {
            int m = v + 8 * half;
            if (mt * 16 + m < M) {
                size_t roff = (size_t)m * DIM;
                hout[roff +  0] = c0[v];
                hout[roff + 16] = c1[v];
                hout[roff + 32] = c2[v];
                hout[roff + 48] = c3[v];
            }
        }
    }
}

// ---------------- out = b + h * dinv^2 (self loop) ----------------

__global__ void init_out_kernel(const float* __restrict__ h,
                                const float* __restrict__ dinv,
                                const float* __restrict__ b,
                                float* __restrict__ out, int n) {
    size_t i = (size_t)blockIdx.x * blockDim.x + threadIdx.x;
    size_t total = (size_t)n * DIM;
    if (i < total) {
        int node = (int)(i >> 9);
        int d    = (int)(i & (DIM - 1));
        float s  = dinv[node];
        out[i] = b[d] + h[i] * s * s;
    }
}

// ---------------- edge scatter: out[col] += h[row] * norm ----------------
// One block per edge, 128 threads x float4 = 512 elems. Atomic f32 adds land
// in L2 (h/out fit in 192MB L2 -> this is L2-resident traffic).

__global__ __launch_bounds__(128) void scatter_kernel(
        const long long* __restrict__ ei,
        const float* __restrict__ ea,
        const float* __restrict__ dinv,
        const float* __restrict__ h,
        float* __restrict__ out, int E) {
    int e = blockIdx.x;
    int r = (int)ei[e];
    int c = (int)ei[(size_t)E + e];
    float norm = dinv[r] * ea[e] * dinv[c];

    const float4* hs = (const float4*)(h + (size_t)r * DIM);
    float*        od = out + (size_t)c * DIM;

    float4 v = hs[threadIdx.x];
    int d = threadIdx.x * 4;
    atomicAdd(&od[d + 0], v.x * norm);
    atomicAdd(&od[d + 1], v.y * norm);
    atomicAdd(&od[d + 2], v.z * norm);
    atomicAdd(&od[d + 3], v.w * norm);
}

// ---------------- launcher ----------------

extern "C" void kernel_launch(void* const* d_in, const int* in_sizes, int n_in,
                              void* d_out, int out_size, void* d_ws, size_t ws_size,
                              hipStream_t stream) {
    const float*     x  = (const float*)d_in[0];
    const long long* ei = (const long long*)d_in[1];   // int64 edge_index [2, E]
    const float*     ea = (const float*)d_in[2];
    const float*     Wm = (const float*)d_in[3];
    const float*     b  = (const float*)d_in[4];
    float*           out = (float*)d_out;

    int N = in_sizes[0] / DIM;
    int E = in_sizes[1] / 2;

    // workspace layout: [deg/dinv: N floats (256-aligned)] [h: N*512 floats]
    float* deg = (float*)d_ws;
    float* h   = deg + ((size_t)(N + 255) & ~(size_t)255);

    deg_init_kernel<<<(N + 255) / 256, 256, 0, stream>>>(deg, N);
    deg_acc_kernel<<<(E + 255) / 256, 256, 0, stream>>>(ei, ea, deg, E);
    dinv_kernel<<<(N + 255) / 256, 256, 0, stream>>>(deg, N);

    int mtiles = (N + 15) / 16;
    int waves  = mtiles * (DIM / 64);          // 4 N-tiles per wave
    gemm_wmma_kernel<<<(waves + 7) / 8, 256, 0, stream>>>(x, Wm, h, N);

    size_t total = (size_t)N * DIM;
    init_out_kernel<<<(int)((total + 255) / 256), 256, 0, stream>>>(h, deg, b, out, N);

    scatter_kernel<<<E, 128, 0, stream>>>(ei, ea, deg, h, out, E);
}